// SimpleGNN_63565515981353
// MI455X (gfx1250) — compile-verified
//
#include <hip/hip_runtime.h>
#include <hip/hip_bf16.h>
#include <math.h>

#define NN   50000
#define EE   800000
#define NG   256
#define F    64

typedef __attribute__((ext_vector_type(2))) float v2f;
typedef __attribute__((ext_vector_type(8))) float v8f;

// ---------------- elementwise helpers ----------------
__global__ void k_zero(float* __restrict__ p, int n) {
    int i = blockIdx.x * blockDim.x + threadIdx.x;
    if (i < n) p[i] = 0.0f;
}

__global__ void k_count(const int* __restrict__ dst, float* __restrict__ deg) {
    int e = blockIdx.x * blockDim.x + threadIdx.x;
    if (e < EE) atomicAdd(&deg[dst[e]], 1.0f);
}

__global__ void k_dinv(const float* __restrict__ deg, float* __restrict__ dinv) {
    int i = blockIdx.x * blockDim.x + threadIdx.x;
    if (i < NN) dinv[i] = rsqrtf(deg[i] + 1.0f);   // deg includes self-loop
}

// agg[i,f] = h[i,f] / deg[i]   (self-loop term; seeds the accumulator)
__global__ void k_self(const float* __restrict__ h, const float* __restrict__ dinv,
                       float* __restrict__ agg) {
    int i = blockIdx.x * blockDim.x + threadIdx.x;   // over NN*F
    if (i < NN * F) {
        int node = i >> 6;
        float di = dinv[node];
        agg[i] = h[i] * di * di;
    }
}

// scatter: agg[dst] += h[src] * dinv[src]*dinv[dst]; 64 lanes per edge
__global__ void k_scatter(const int* __restrict__ src, const int* __restrict__ dst,
                          const float* __restrict__ h, const float* __restrict__ dinv,
                          float* __restrict__ agg) {
    int t = blockIdx.x * 256 + threadIdx.x;          // EE*64 threads total
    int e = t >> 6;
    int f = t & 63;
    if (e < EE) {
        int s = src[e];
        int d = dst[e];
        float norm = dinv[s] * dinv[d];
        atomicAdd(&agg[(size_t)d * F + f], h[(size_t)s * F + f] * norm);
    }
}

__global__ void k_bias_relu(float* __restrict__ agg, const float* __restrict__ b) {
    int i = blockIdx.x * blockDim.x + threadIdx.x;
    if (i < NN * F) {
        float v = agg[i] + b[i & 63];
        agg[i] = v > 0.0f ? v : 0.0f;
    }
}

__global__ void k_pool(const float* __restrict__ agg, const int* __restrict__ batch,
                       float* __restrict__ pooled, float* __restrict__ cnt) {
    int i = blockIdx.x * blockDim.x + threadIdx.x;   // over NN*F
    if (i < NN * F) {
        int node = i >> 6;
        int f = i & 63;
        int g = batch[node];
        atomicAdd(&pooled[g * F + f], agg[i]);
        if (f == 0) atomicAdd(&cnt[g], 1.0f);
    }
}

__global__ void k_head(const float* __restrict__ pooled, const float* __restrict__ cnt,
                       const float* __restrict__ Wl, const float* __restrict__ bl,
                       float* __restrict__ out) {
    int g = threadIdx.x;
    if (g < NG) {
        float s = 0.0f;
        #pragma unroll
        for (int f = 0; f < F; ++f) s += pooled[g * F + f] * Wl[f];
        float c = cnt[g];
        c = c > 1.0f ? c : 1.0f;
        out[g] = s / c + bl[0];
    }
}

// ---------------- WMMA fp32 GEMM: out[M,64] = in[M,64] @ W[64,64] ----------------
// One wave computes a 16x64 output tile (4 accumulators), 4 waves per block.
// f32 16x16x4 operand layouts per CDNA5 ISA 7.12.2:
//   A 16x4 : lane m=lane&15; VGPR0=A[m][2*half+0], VGPR1=A[m][2*half+1]
//   B 4x16 : lane n=lane&15; VGPR0=B[2*half+0][n], VGPR1=B[2*half+1][n]
//   C/D    : VGPR r -> (row = r + 8*half, col = lane&15)
__global__ void __launch_bounds__(128)
gemm64_wmma(const float* __restrict__ in, const float* __restrict__ W,
            float* __restrict__ out, int nrows) {
    int wave = threadIdx.x >> 5;
    int lane = threadIdx.x & 31;
    int tile = blockIdx.x * 4 + wave;            // 16-row tile (wave-uniform)
    if (tile * 16 >= nrows) return;              // whole wave exits: EXEC stays full
    int row0 = tile * 16;
    int half = lane >> 4;
    int lm   = lane & 15;

    v8f acc0 = {}, acc1 = {}, acc2 = {}, acc3 = {};
    const float* arow = in + (size_t)(row0 + lm) * F;

    #pragma unroll
    for (int kt = 0; kt < 16; ++kt) {
        int k0 = kt * 4 + 2 * half;
        v2f a;
        a.x = arow[k0];
        a.y = arow[k0 + 1];
        const float* wr0 = W + (size_t)k0 * F + lm;   // row k0   of W
        const float* wr1 = wr0 + F;                   // row k0+1 of W
        v2f b0, b1, b2, b3;
        b0.x = wr0[0];   b0.y = wr1[0];
        b1.x = wr0[16];  b1.y = wr1[16];
        b2.x = wr0[32];  b2.y = wr1[32];
        b3.x = wr0[48];  b3.y = wr1[48];
        acc0 = __builtin_amdgcn_wmma_f32_16x16x4_f32(false, a, false, b0, (short)0, acc0, false, false);
        acc1 = __builtin_amdgcn_wmma_f32_16x16x4_f32(false, a, false, b1, (short)0, acc1, false, false);
        acc2 = __builtin_amdgcn_wmma_f32_16x16x4_f32(false, a, false, b2, (short)0, acc2, false, false);
        acc3 = __builtin_amdgcn_wmma_f32_16x16x4_f32(false, a, false, b3, (short)0, acc3, false, false);
    }

    float* orow = out + (size_t)row0 * F;
    #pragma unroll
    for (int r = 0; r < 8; ++r) {
        int row = r + 8 * half;
        orow[(size_t)row * F +  0 + lm] = acc0[r];
        orow[(size_t)row * F + 16 + lm] = acc1[r];
        orow[(size_t)row * F + 32 + lm] = acc2[r];
        orow[(size_t)row * F + 48 + lm] = acc3[r];
    }
}

// ---------------- launcher ----------------
extern "C" void kernel_launch(void* const* d_in, const int* in_sizes, int n_in,
                              void* d_out, int out_size, void* d_ws, size_t ws_size,
                              hipStream_t stream) {
    (void)in_sizes; (void)n_in; (void)out_size; (void)ws_size;

    const float* x     = (const float*)d_in[0];
    const int*   src   = (const int*)  d_in[1];
    const int*   dst   = (const int*)  d_in[2];
    const int*   batch = (const int*)  d_in[3];
    const float* W1 = (const float*)d_in[4];  const float* b1 = (const float*)d_in[5];
    const float* W2 = (const float*)d_in[6];  const float* b2 = (const float*)d_in[7];
    const float* W3 = (const float*)d_in[8];  const float* b3 = (const float*)d_in[9];
    const float* Wl = (const float*)d_in[10]; const float* bl = (const float*)d_in[11];
    float* out = (float*)d_out;

    // workspace carve-up (floats)
    float* ws     = (float*)d_ws;
    float* deg    = ws;                       // NN
    float* dinv   = deg + NN;                 // NN
    float* h      = dinv + NN;                // NN*F
    float* agg    = h + (size_t)NN * F;       // NN*F
    float* pooled = agg + (size_t)NN * F;     // NG*F
    float* cnt    = pooled + NG * F;          // NG   (contiguous with pooled)

    const int TPB = 256;
    int gN   = (NN + TPB - 1) / TPB;            // over nodes
    int gNF  = (NN * F) / TPB;                  // 3.2M / 256 = 12500 exact
    int gE   = (EE + TPB - 1) / TPB;            // over edges
    int gEF  = (EE * F) / TPB;                  // 200000 exact
    int gGEM = ((NN / 16) + 3) / 4;             // 3125 tiles, 4 waves/block -> 782

    // degrees (once)
    k_zero <<<gN, TPB, 0, stream>>>(deg, NN);
    k_zero <<<(NG * F + NG + TPB - 1) / TPB, TPB, 0, stream>>>(pooled, NG * F + NG);
    k_count<<<gE, TPB, 0, stream>>>(dst, deg);
    k_dinv <<<gN, TPB, 0, stream>>>(deg, dinv);

    // layer 1
    gemm64_wmma<<<gGEM, 128, 0, stream>>>(x, W1, h, NN);
    k_self     <<<gNF, TPB, 0, stream>>>(h, dinv, agg);
    k_scatter  <<<gEF, TPB, 0, stream>>>(src, dst, h, dinv, agg);
    k_bias_relu<<<gNF, TPB, 0, stream>>>(agg, b1);

    // layer 2 (gemm reads agg before k_self overwrites it: stream-ordered)
    gemm64_wmma<<<gGEM, 128, 0, stream>>>(agg, W2, h, NN);
    k_self     <<<gNF, TPB, 0, stream>>>(h, dinv, agg);
    k_scatter  <<<gEF, TPB, 0, stream>>>(src, dst, h, dinv, agg);
    k_bias_relu<<<gNF, TPB, 0, stream>>>(agg, b2);

    // layer 3
    gemm64_wmma<<<gGEM, 128, 0, stream>>>(agg, W3, h, NN);
    k_self     <<<gNF, TPB, 0, stream>>>(h, dinv, agg);
    k_scatter  <<<gEF, TPB, 0, stream>>>(src, dst, h, dinv, agg);
    k_bias_relu<<<gNF, TPB, 0, stream>>>(agg, b3);

    // pool + head
    k_pool<<<gNF, TPB, 0, stream>>>(agg, batch, pooled, cnt);
    k_head<<<1, TPB, 0, stream>>>(pooled, cnt, Wl, bl, out);
}